// GraphAttentionLayer_36825049596096
// MI455X (gfx1250) — compile-verified
//
#include <hip/hip_runtime.h>
#include <hip/hip_bf16.h>
#include <math.h>

// Shapes from the reference
#define BS 2
#define NV 8
#define MF 16
#define MT 16
#define NPTH 4
#define KH 3
#define NR 12
#define HD 32
#define OUT 32
// Flattened sizes
#define NM (BS*NV*MF*MT*NPTH)   // 16384 path rows
#define NBVF (BS*NV*MF)         // 256
#define NBV (BS*NV)             // 16

typedef __attribute__((ext_vector_type(2))) float v2f;
typedef __attribute__((ext_vector_type(8))) float v8f;

// -------------------------------------------------------------------------
// K1: M_j5[m,d] = (prod_k  sum_r rel[m,k,r]*diagW[r,d]) * fe[bvf(m),d]
// One wave per 16-row x 32-col tile. Pure-f32 WMMA: V_WMMA_F32_16X16X4_F32.
// K=12 -> 3 chunks of 4; N=32 -> 2 tiles; 3 hops kept in separate
// accumulators (the hop combination is an elementwise product, not a sum).
// A layout (ISA 7.12.2, 32-bit A 16x4): lane L holds row M=L&15;
//   lanes 0-15 carry K={0,1} in {v0,v1}, lanes 16-31 carry K={2,3}.
// B layout (4x16, rows striped across lanes per VGPR, mirroring C/D):
//   VGPR j, lanes 0-15: K=j, N=lane; lanes 16-31: K=j+2, N=lane-16.
// C/D layout: VGPR j, lanes 0-15: (M=j, N=lane); lanes 16-31: (M=j+8).
// -------------------------------------------------------------------------
__global__ __launch_bounds__(32)
void k1_wmma_diag(const float* __restrict__ rel,   // [NM, KH, NR]
                  const float* __restrict__ Wfull, // [NR, HD, HD]
                  const float* __restrict__ fe,    // [BS,NV,MF,HD]
                  float* __restrict__ mj5)         // [NM, HD]
{
    const int lane = threadIdx.x;
    const int m0   = blockIdx.x * 16;
    const int hi   = lane >> 4;   // 0/1 half of the wave
    const int lo   = lane & 15;

    // B tiles: diagW[r,d] = Wfull[r*HD*HD + d*(HD+1)]
    v2f Bt[3][2];
#pragma unroll
    for (int kc = 0; kc < 3; ++kc) {
#pragma unroll
        for (int nh = 0; nh < 2; ++nh) {
            const int d = nh * 16 + lo;
            Bt[kc][nh].x = Wfull[(kc * 4 + 0 + 2 * hi) * (HD * HD) + d * (HD + 1)];
            Bt[kc][nh].y = Wfull[(kc * 4 + 1 + 2 * hi) * (HD * HD) + d * (HD + 1)];
        }
    }

    v8f acc[3][2];
#pragma unroll
    for (int k = 0; k < 3; ++k)
#pragma unroll
        for (int nh = 0; nh < 2; ++nh)
            acc[k][nh] = v8f{0.f,0.f,0.f,0.f,0.f,0.f,0.f,0.f};

    const float* arow = rel + (size_t)(m0 + lo) * (KH * NR);
#pragma unroll
    for (int k = 0; k < 3; ++k) {           // relation hops (elementwise-combined later)
#pragma unroll
        for (int kc = 0; kc < 3; ++kc) {    // K chunks of 4 over the 12 relations
            v2f A;
            A.x = arow[k * NR + kc * 4 + 2 * hi + 0];
            A.y = arow[k * NR + kc * 4 + 2 * hi + 1];
#pragma unroll
            for (int nh = 0; nh < 2; ++nh) {
                acc[k][nh] = __builtin_amdgcn_wmma_f32_16x16x4_f32(
                    /*neg_a=*/false, A, /*neg_b=*/false, Bt[kc][nh],
                    /*c_mod=*/(short)0, acc[k][nh],
                    /*reuse_a=*/false, /*reuse_b=*/false);
            }
        }
    }

    // Elementwise hop product * feature embedding, scatter back per C/D layout
#pragma unroll
    for (int nh = 0; nh < 2; ++nh) {
        const int d = nh * 16 + lo;
#pragma unroll
        for (int j = 0; j < 8; ++j) {
            const int m = m0 + j + 8 * hi;
            float p = acc[0][nh][j] * acc[1][nh][j] * acc[2][nh][j];
            p *= fe[(m >> 6) * HD + d];       // bvf = m / (MT*NPTH)
            mj5[(size_t)m * HD + d] = p;
        }
    }
}

// -------------------------------------------------------------------------
// K2: per (b,v,f): Gumbel-softmax path gating + path_attn + reduce over t,p.
// 64 threads = (t=tid>>2, p=tid&3). Deterministic serial LDS reduction.
// -------------------------------------------------------------------------
__global__ __launch_bounds__(64)
void k2_path(const float* __restrict__ mj5, const float* __restrict__ h_time,
             const float* __restrict__ gum, const float* __restrict__ W_p,
             const float* __restrict__ pmw, const float* __restrict__ pmb,
             float* __restrict__ mjred,         // [NBVF, HD]
             float* __restrict__ out_path_attn) // [NM]
{
    __shared__ float red[64 * 33];
    const int bvf = blockIdx.x;
    const int tid = threadIdx.x;
    const int b = bvf >> 7;           // / (NV*MF)
    const int v = (bvf >> 4) & 7;
    const int m = bvf * 64 + tid;     // t*4+p == tid
    const int p = tid & 3;

    float vec[HD];
    const float* src = mj5 + (size_t)m * HD;
    float msgdot = 0.f, pmdot = 0.f;
#pragma unroll
    for (int d = 0; d < HD; ++d) {
        vec[d] = src[d];
        msgdot += vec[d] * W_p[OUT + d];
        pmdot  += vec[d] * pmw[d];
    }
    const float* ht = h_time + (b * NV + v) * OUT;
    float htdot = 0.f;
#pragma unroll
    for (int d = 0; d < OUT; ++d) htdot += ht[d] * W_p[d];

    const float logit = htdot + msgdot + gum[m];

    // softmax / argmax over the 4 path lanes (groups of 4 within a wave)
    float mx = logit;
    mx = fmaxf(mx, __shfl_xor(mx, 1, 4));
    mx = fmaxf(mx, __shfl_xor(mx, 2, 4));
    float e = expf(logit - mx);
    float se = e;
    se += __shfl_xor(se, 1, 4);
    se += __shfl_xor(se, 2, 4);
    const float ysoft = e / se;

    const int ismax = (logit == mx) ? 1 : 0;
    const int f0 = __shfl(ismax, 0, 4);
    const int f1 = __shfl(ismax, 1, 4);
    const int f2 = __shfl(ismax, 2, 4);
    const int amax = f0 ? 0 : (f1 ? 1 : (f2 ? 2 : 3)); // first-index tiebreak
    const float yhard = (p == amax) ? 1.f : 0.f;
    const float mask = (yhard + ysoft) - ysoft;        // straight-through, literal

    // path_score . path_mlp_w == mask * (vec . pmw) since mask is scalar
    const float pa = 1.f / (1.f + expf(-(mask * pmdot + pmb[0])));
    out_path_attn[m] = pa;

#pragma unroll
    for (int d = 0; d < HD; ++d) red[tid * 33 + d] = pa * vec[d];
    __syncthreads();
    if (tid < HD) {
        float s = 0.f;
        for (int i = 0; i < 64; ++i) s += red[i * 33 + tid]; // fixed order
        mjred[bvf * HD + tid] = s;
    }
}

// -------------------------------------------------------------------------
// K3: per (b,v): joint_impact (view-reinterpretation cos diagonal) + causal.
// 32 threads, thread == channel d.
// -------------------------------------------------------------------------
__global__ __launch_bounds__(32)
void k3_joint(const float* __restrict__ mjred, const float* __restrict__ h_time,
              const float* __restrict__ cmw, const float* __restrict__ cmb,
              const float* __restrict__ cth,
              float* __restrict__ out_gc, float* __restrict__ out_gt,
              float* __restrict__ out_ac)
{
    __shared__ float ne_s[MF * HD];   // 512: flat per (b,v) for the .view trick
    __shared__ float cosd[MF];
    __shared__ float ac_s[MF];
    const int bv = blockIdx.x;
    const int d = threadIdx.x;
    const float* M = mjred + bv * (MF * HD);

    for (int f = 0; f < MF; ++f) {
        const float val = M[f * HD + d];
        float s = val * val;
#pragma unroll
        for (int off = 16; off >= 1; off >>= 1) s += __shfl_xor(s, off, 32);
        const float norm = sqrtf(s);
        ne_s[f * HD + d] = val / fmaxf(norm, 1e-12f);
    }
    __syncthreads();

    if (d < MF) {
        const int f = d;
        float c = 0.f;
        for (int h = 0; h < HD; ++h)
            c += ne_s[f * HD + h] * ne_s[h * MF + f]; // n2 = row-major reinterpretation
        cosd[f] = c;
    }
    __syncthreads();

    const float* ht = h_time + bv * OUT;
    if (d < MF) {
        const int f = d;
        float sc = cmb[0];
        for (int j = 0; j < OUT; ++j) sc += ht[j] * cmw[j];
        const float cd = cosd[f];
        for (int j = 0; j < HD; ++j) sc += cd * M[f * HD + j] * cmw[OUT + j];
        const float ac = 1.f / (1.f + expf(-(sc - cth[0])));
        ac_s[f] = ac;
        out_ac[bv * MF + f] = ac;
    }
    __syncthreads();

    float gc = 0.f, gt = 0.f;
    for (int f = 0; f < MF; ++f) {
        const float mn = cosd[f] * M[f * HD + d];
        gc += ac_s[f] * mn;
        gt += (1.f - ac_s[f]) * mn;
    }
    out_gc[bv * HD + d] = gc;
    out_gt[bv * HD + d] = gt;
}

// -------------------------------------------------------------------------
// K4: g_i = g_c + g_t[II] with II = clip(perm + (perm==arange(bs)), 0, bs-1)
// -------------------------------------------------------------------------
__global__ void k4_gi(const float* __restrict__ gc, const float* __restrict__ gt,
                      const int* __restrict__ perm, float* __restrict__ gi)
{
    const int idx = blockIdx.x * blockDim.x + threadIdx.x;
    if (idx >= BS * NV * HD) return;
    const int b = idx / (NV * HD);
    const int rest = idx % (NV * HD);
    int II = perm[b] + ((perm[b] == b) ? 1 : 0);
    II = II < 0 ? 0 : (II > BS - 1 ? BS - 1 : II);
    gi[idx] = gc[idx] + gt[II * (NV * HD) + rest];
}

extern "C" void kernel_launch(void* const* d_in, const int* in_sizes, int n_in,
                              void* d_out, int out_size, void* d_ws, size_t ws_size,
                              hipStream_t stream)
{
    const float* fe   = (const float*)d_in[0];   // feature_embed
    const float* ht   = (const float*)d_in[1];   // h_time
    const float* rel  = (const float*)d_in[2];   // rel_index
    const float* gum  = (const float*)d_in[3];   // gumbel_noise
    const float* W    = (const float*)d_in[4];   // params_W
    const float* W_p  = (const float*)d_in[5];
    const float* pmw  = (const float*)d_in[6];   // path_mlp_w
    const float* pmb  = (const float*)d_in[7];   // path_mlp_b (1 elem)
    const float* cmw  = (const float*)d_in[8];   // causal_mlp_w
    const float* cmb  = (const float*)d_in[9];   // causal_mlp_b (1 elem)
    const float* cth  = (const float*)d_in[10];  // causal_threshold (1 elem)
    const int*   perm = (const int*)d_in[11];    // perm_idx

    float* out   = (float*)d_out;
    float* gi    = out;                          // 512
    float* gc    = out + 512;                    // 512
    float* gt    = out + 1024;                   // 512
    float* pattn = out + 1536;                   // 16384
    float* ac    = out + 1536 + NM;              // 256

    float* mj5   = (float*)d_ws;                 // NM*HD  = 524288 floats (2 MB)
    float* mjred = mj5 + (size_t)NM * HD;        // NBVF*HD = 8192 floats

    k1_wmma_diag<<<NM / 16, 32, 0, stream>>>(rel, W, fe, mj5);
    k2_path<<<NBVF, 64, 0, stream>>>(mj5, ht, gum, W_p, pmw, pmb, mjred, pattn);
    k3_joint<<<NBV, 32, 0, stream>>>(mjred, ht, cmw, cmb, cth, gc, gt, ac);
    k4_gi<<<2, 256, 0, stream>>>(gc, gt, perm, gi);
}